// PrunedGroupSum_86277303042369
// MI455X (gfx1250) — compile-verified
//
#include <hip/hip_runtime.h>
#include <hip/hip_bf16.h>

// PrunedGroupSum for MI455X (gfx1250): y[b,k] = (sum_{d in group k} x[b,d]) / TAU + BETA
// Bandwidth-bound (1 GiB stream, ~46us floor at 23.3 TB/s). WMMA f32 16x16x4 does the
// segmented scatter-accumulate: A = 16 rows x 4 cols of x, B = 4x16 one-hot col->group
// selection, C = 16x16 fp32 accumulator (16 rows x 16-group sliding window).

#define TAU_F 20.0f
#define BETA_F 0.0f
#define CHUNK 512          // columns per block tile
#define ROWS 16            // batch rows per block tile
#define LDS_STRIDE 516     // floats per LDS row: 516 mod 64 == 4 -> conflict-free b64 reads

typedef float v2f __attribute__((ext_vector_type(2)));
typedef float v8f __attribute__((ext_vector_type(8)));

// ---------------- init output to BETA ----------------
__global__ void k_zero(float* __restrict__ out, int n) {
  int i = blockIdx.x * blockDim.x + threadIdx.x;
  if (i < n) out[i] = BETA_F;
}

// ---------------- inclusive prefix sum of group_sizes (K <= 1024) ----------------
__global__ void k_prefix(const int* __restrict__ gs, int* __restrict__ offsets, int K) {
  __shared__ int s[1024];
  int t = threadIdx.x;
  s[t] = (t < K) ? gs[t] : 0;
  __syncthreads();
  for (int off = 1; off < 1024; off <<= 1) {
    int v = (t >= off) ? s[t - off] : 0;
    __syncthreads();
    s[t] += v;
    __syncthreads();
  }
  if (t < K) offsets[t] = s[t];
}

// ---------------- column -> group id via upper_bound on cumsum ----------------
__global__ void k_seg(const int* __restrict__ offsets, int* __restrict__ grp, int D, int K) {
  int c = blockIdx.x * blockDim.x + threadIdx.x;
  if (c >= D) return;
  int lo = 0, hi = K;
  while (lo < hi) {
    int mid = (lo + hi) >> 1;
    if (offsets[mid] <= c) lo = mid + 1; else hi = mid;
  }
  grp[c] = lo;
}

// Flush the 16x16 accumulator window: lane holds column slot n = lane&15 for rows
// M = half*8 + j (C/D layout: VGPR j -> M=j (lanes 0-15) / M=j+8 (lanes 16-31)).
static __device__ inline void flush_acc(const v8f& acc, float* __restrict__ out,
                                        int row0, int half, int n, int g0, int nUsed, int K) {
  if (n < nUsed) {
    float* p = out + (size_t)(row0 + half * 8) * (size_t)K + (size_t)(g0 + n);
#pragma unroll
    for (int j = 0; j < 8; ++j)
      atomicAdd(p + (size_t)j * (size_t)K, acc[j] * (1.0f / TAU_F));
  }
}

// ---------------- main: async tile load + WMMA segmented reduce ----------------
__global__ void __launch_bounds__(256) k_main(const float* __restrict__ x,
                                              const int* __restrict__ grp,
                                              float* __restrict__ out,
                                              int D, int K) {
  __shared__ __align__(16) float tile[ROWS * LDS_STRIDE];  // 33 KB x-tile
  __shared__ __align__(16) int   lgrp[CHUNK];              // 2 KB group ids

  const int tid  = threadIdx.x;
  const int c0   = blockIdx.x * CHUNK;
  const int row0 = blockIdx.y * ROWS;

  // ----- Stage 16x512 fp32 x-tile + 512 group-ids into LDS with async b128 copies -----
  // x-tile: 2048 float4s; thread t of iteration i moves float4 (i*256+t):
  //   row = f>>7, col4 = f&127 -> 128 consecutive threads stream 2 KB contiguous.
  const float* gbase = x + (size_t)row0 * (size_t)D + (size_t)c0;
#pragma unroll
  for (int i = 0; i < 8; ++i) {
    int f  = i * 256 + tid;
    int r  = f >> 7;
    int c4 = f & 127;
    unsigned goff  = (unsigned)((r * D + c4 * 4) * 4);
    unsigned laddr = (unsigned)(unsigned long long)(tile + r * LDS_STRIDE + c4 * 4);
    asm volatile("global_load_async_to_lds_b128 %0, %1, %2"
                 :: "v"(laddr), "v"(goff), "s"(gbase) : "memory");
  }
  // group ids: 512 ints = 128 int4s (reused by all 8 waves, L2-hot source)
  if (tid < 128) {
    const int* grpbase = grp + c0;
    unsigned goff  = (unsigned)(tid * 16);
    unsigned laddr = (unsigned)(unsigned long long)(lgrp + tid * 4);
    asm volatile("global_load_async_to_lds_b128 %0, %1, %2"
                 :: "v"(laddr), "v"(goff), "s"(grpbase) : "memory");
  }
  asm volatile("s_wait_asynccnt 0" ::: "memory");
  __syncthreads();

  // ----- Per-wave WMMA segmented accumulate over 64 columns -----
  const int lane    = tid & 31;
  const int wv      = tid >> 5;        // 8 waves, 64 cols each
  const int colBase = wv * 64;
  const int n    = lane & 15;          // B/C column slot
  const int half = lane >> 4;          // 0: K=0,1 / 1: K=2,3 (A and B fragments)
  const int m    = lane & 15;          // A row

  v8f acc = {};
  int g0   = lgrp[colBase];            // base of 16-group window
  int gmax = g0;

  const float* lrow = tile + m * LDS_STRIDE;

  // 4 units of 16 columns. 16 consecutive columns span at most 16 groups, so the
  // window-overflow check runs once per unit; the 4 WMMA steps inside are branch-free.
  for (int u = 0; u < 4; ++u) {
    const int cU = colBase + u * 16;
    const int gfirst = lgrp[cU];
    const int glast  = lgrp[cU + 15];

    if (glast - g0 >= 16) {            // rare: flush window and re-base
      flush_acc(acc, out, row0, half, n, g0, gmax - g0 + 1, K);
      acc = {};
      g0 = gfirst;
    }
    gmax = glast;

#pragma unroll
    for (int s = 0; s < 4; ++s) {
      const int cLoc = cU + s * 4;
      const int4 g4 = *(const int4*)(lgrp + cLoc);   // LDS broadcast read

      // A fragment (16x4 f32): lane m, VGPR0 K=2*half, VGPR1 K=2*half+1 -> ds_load_b64,
      // stride 516 words => 64 distinct banks across the wave.
      const float2 av = *(const float2*)(lrow + cLoc + half * 2);
      v2f a; a.x = av.x; a.y = av.y;

      // B fragment (4x16 0/1 selection): row K=2*half (+1), col n; B[k][n]=1 iff
      // column (c+k) belongs to local group n.
      int ga = half ? g4.z : g4.x;
      int gb = half ? g4.w : g4.y;
      v2f b;
      b.x = (ga - g0 == n) ? 1.0f : 0.0f;
      b.y = (gb - g0 == n) ? 1.0f : 0.0f;

      acc = __builtin_amdgcn_wmma_f32_16x16x4_f32(
          /*neg_a=*/false, a, /*neg_b=*/false, b,
          /*c_mod=*/(short)0, acc, /*reuse_a=*/false, /*reuse_b=*/false);
    }
  }

  flush_acc(acc, out, row0, half, n, g0, gmax - g0 + 1, K);
}

// ---------------- host launch ----------------
extern "C" void kernel_launch(void* const* d_in, const int* in_sizes, int n_in,
                              void* d_out, int out_size, void* d_ws, size_t ws_size,
                              hipStream_t stream) {
  const float* x  = (const float*)d_in[0];
  const int*   gs = (const int*)d_in[1];
  float* out = (float*)d_out;

  const int K = in_sizes[1];
  const int B = out_size / K;
  const int D = in_sizes[0] / B;

  int* offsets = (int*)d_ws;          // K ints (padded to 1024)
  int* grp     = (int*)d_ws + 1024;   // D ints

  k_zero<<<(out_size + 255) / 256, 256, 0, stream>>>(out, out_size);
  k_prefix<<<1, 1024, 0, stream>>>(gs, offsets, K);
  k_seg<<<(D + 255) / 256, 256, 0, stream>>>(offsets, grp, D, K);

  dim3 grid(D / CHUNK, B / ROWS);
  k_main<<<grid, 256, 0, stream>>>(x, grp, out, D, K);
}